// RobotLayer_89034672046885
// MI455X (gfx1250) — compile-verified
//
#include <hip/hip_runtime.h>
#include <hip/hip_bf16.h>
#include <stdint.h>

// Forward kinematics, 9-link chain, 7 revolute joints (axes z,y,z,y,z,y,z).
// Bandwidth-bound (≈334 MB traffic, 288 MB of it output stores → ~14 µs floor
// at 23.3 TB/s). Strategy: per-thread register chain, stage the block's output
// tile in LDS, drain it with one TDM tensor_store_from_lds per workgroup.

#define BATCH_PER_BLOCK 64
#define NLINK 9
#define ROW_F 144          // floats per element output: 9 links * 16
#define TILE_BYTES (BATCH_PER_BLOCK * ROW_F * 4)   // 36 KB LDS per block

typedef uint32_t u32x4 __attribute__((ext_vector_type(4)));
typedef int      i32x4 __attribute__((ext_vector_type(4)));
typedef int      i32x8 __attribute__((ext_vector_type(8)));

__device__ __forceinline__ void mat4_mul_rhs(const float* __restrict__ A,
                                             const float* __restrict__ Bm,
                                             float* __restrict__ C) {
  // C = A(4x4) * Bm(4x4), row-major
#pragma unroll
  for (int r = 0; r < 4; ++r) {
#pragma unroll
    for (int c = 0; c < 4; ++c) {
      C[r * 4 + c] = A[r * 4 + 0] * Bm[0 * 4 + c]
                   + A[r * 4 + 1] * Bm[1 * 4 + c]
                   + A[r * 4 + 2] * Bm[2 * 4 + c]
                   + A[r * 4 + 3] * Bm[3 * 4 + c];
    }
  }
}

__global__ __launch_bounds__(BATCH_PER_BLOCK)
void fk_chain_kernel(const float* __restrict__ Tbase,
                     const float* __restrict__ Toff,
                     const float* __restrict__ Q,
                     float* __restrict__ out,
                     int B) {
  __shared__ float smem[BATCH_PER_BLOCK * ROW_F];   // exact flat output tile

  const int tid = threadIdx.x;
  const int b   = blockIdx.x * BATCH_PER_BLOCK + tid;

  if (b < B) {
    // ---- load state ----
    float T[16];
    const float4* tb4 = reinterpret_cast<const float4*>(Tbase + (size_t)b * 16);
#pragma unroll
    for (int i = 0; i < 4; ++i) {
      float4 v = tb4[i];
      T[i * 4 + 0] = v.x; T[i * 4 + 1] = v.y; T[i * 4 + 2] = v.z; T[i * 4 + 3] = v.w;
    }
    float q[7];
    const float* qp = Q + (size_t)b * 7;
#pragma unroll
    for (int i = 0; i < 7; ++i) q[i] = qp[i];

    // ---- chain: axes per link (0 = fixed, 2 = y, 3 = z) ----
    constexpr int axes[NLINK] = {0, 3, 2, 3, 2, 3, 2, 3, 0};
    int iq = 0;
    float4* dst = reinterpret_cast<float4*>(&smem[tid * ROW_F]);

#pragma unroll
    for (int i = 0; i < NLINK; ++i) {
      float Tn[16];
      mat4_mul_rhs(T, Toff + i * 16, Tn);   // Toff thread-uniform -> scalar loads

      if (axes[i] == 3) {                   // right-mult by Rz(q)
        float s, c;
        __sincosf(q[iq], &s, &c);
        iq++;
#pragma unroll
        for (int r = 0; r < 4; ++r) {
          float t0 = Tn[r * 4 + 0], t1 = Tn[r * 4 + 1];
          Tn[r * 4 + 0] = c * t0 + s * t1;
          Tn[r * 4 + 1] = c * t1 - s * t0;
        }
      } else if (axes[i] == 2) {            // right-mult by Ry(q)
        float s, c;
        __sincosf(q[iq], &s, &c);
        iq++;
#pragma unroll
        for (int r = 0; r < 4; ++r) {
          float t0 = Tn[r * 4 + 0], t2 = Tn[r * 4 + 2];
          Tn[r * 4 + 0] = c * t0 - s * t2;
          Tn[r * 4 + 2] = s * t0 + c * t2;
        }
      }

#pragma unroll
      for (int k = 0; k < 16; ++k) T[k] = Tn[k];

      // stage this link's transform in LDS (float4 -> ds_store_b128)
      dst[i * 4 + 0] = make_float4(T[0],  T[1],  T[2],  T[3]);
      dst[i * 4 + 1] = make_float4(T[4],  T[5],  T[6],  T[7]);
      dst[i * 4 + 2] = make_float4(T[8],  T[9],  T[10], T[11]);
      dst[i * 4 + 3] = make_float4(T[12], T[13], T[14], T[15]);
    }
  }

  __syncthreads();

  // valid rows in this block's tile (tail block may be partial)
  int rows = B - blockIdx.x * BATCH_PER_BLOCK;
  if (rows > BATCH_PER_BLOCK) rows = BATCH_PER_BLOCK;

#if defined(__gfx1250__) && __has_builtin(__builtin_amdgcn_tensor_store_from_lds)
  // ---- TDM drain: one linear DMA of the whole tile, issued by wave 0 ----
  if (tid < 32) {
    uint64_t gaddr = (uint64_t)(uintptr_t)out
                   + (uint64_t)blockIdx.x * (uint64_t)TILE_BYTES;
    uint32_t lbase = (uint32_t)(uintptr_t)(&smem[0]);   // LDS byte offset

    // D# group 0: count=1 | lds_addr | global_addr[56:0] | type=2
    u32x4 g0;
    g0[0] = 1u;                                         // count = 1 valid descriptor
    g0[1] = lbase;                                      // lds_addr
    g0[2] = (uint32_t)gaddr;                            // global_addr[31:0]
    g0[3] = ((uint32_t)(gaddr >> 32) & 0x01FFFFFFu)     // global_addr[56:32]
          | (2u << 30);                                 // type = 2 ("image")

    // D# group 1: data_size=4B; 2-D tile: dim0=144 floats, dim1=rows
    const uint32_t td0 = ROW_F;     // tensor_dim0 (elements)
    const uint32_t td1 = (uint32_t)rows;
    i32x8 g1;
    g1[0] = (int)(2u << 16);                            // data_size = 4 bytes
    g1[1] = (int)((td0 & 0xFFFFu) << 16);               // tensor_dim0[15:0]
    g1[2] = (int)(((td0 >> 16) & 0xFFFFu)
          | ((td1 & 0xFFFFu) << 16));                   // td0[31:16] | td1[15:0]
    g1[3] = (int)(((td1 >> 16) & 0xFFFFu)
          | ((uint32_t)ROW_F << 16));                   // td1[31:16] | tile_dim0
    g1[4] = (int)(td1 & 0xFFFFu);                       // tile_dim1 = rows, tile_dim2 = 0
    g1[5] = (int)(uint32_t)ROW_F;                       // tensor_dim0_stride[31:0]
    g1[6] = 0;                                          // stride0[47:32] | stride1[15:0]
    g1[7] = 0;                                          // stride1[47:16]

    i32x4 gz = {0, 0, 0, 0};                            // groups 2/3 unused (2-D)
    i32x8 g4z = {0, 0, 0, 0, 0, 0, 0, 0};               // VADDR4 slot: unused/NULL

    __builtin_amdgcn_tensor_store_from_lds(g0, g1, gz, gz, g4z, 0);
    __builtin_amdgcn_s_wait_tensorcnt(0);               // LDS stays valid until DMA done
  }
#else
  // Fallback: coalesced LDS -> global copy
  {
    size_t gbase = (size_t)blockIdx.x * (size_t)(BATCH_PER_BLOCK * ROW_F);
    int total = rows * ROW_F;
    for (int k = tid; k < total; k += BATCH_PER_BLOCK) out[gbase + k] = smem[k];
  }
#endif
}

extern "C" void kernel_launch(void* const* d_in, const int* in_sizes, int n_in,
                              void* d_out, int out_size, void* d_ws, size_t ws_size,
                              hipStream_t stream) {
  (void)n_in; (void)out_size; (void)d_ws; (void)ws_size;
  const float* Tbase = (const float*)d_in[0];   // [B,4,4]
  const float* Toff  = (const float*)d_in[1];   // [9,4,4]
  const float* Q     = (const float*)d_in[2];   // [B,7]
  float* out = (float*)d_out;                   // [B,9,4,4]

  const int B = in_sizes[0] / 16;
  const int grid = (B + BATCH_PER_BLOCK - 1) / BATCH_PER_BLOCK;
  fk_chain_kernel<<<grid, BATCH_PER_BLOCK, 0, stream>>>(Tbase, Toff, Q, out, B);
}